// GAT_22909355556935
// MI455X (gfx1250) — compile-verified
//
#include <hip/hip_runtime.h>
#include <hip/hip_bf16.h>
#include <math.h>

// GAT layer, MI455X (gfx1250), wave32, WMMA f32 16x16x4 (exact f32 math).
// Fused flash-style masked softmax + PV matmul: the [H,N,N] attention matrix
// (302 MB in f32) never touches memory.

#define GN      3072      // nodes
#define GIN     512       // in_dim
#define GH      8         // heads
#define GO      64        // out_dim per head
#define GALPHA  0.2f
#define GNEGINF (-9.0e15f)

typedef float v2f __attribute__((ext_vector_type(2)));
typedef float v8f __attribute__((ext_vector_type(8)));

// -------------------------------------------------------------------------
// Kernel 1: Wh[h] = x @ W[h]   ([N,512] x [512,64] -> [H,N,64])
// 256 threads = 8 waves; wave w owns a 16x16 output tile:
//   rows m0 = blockIdx.x*32 + (w>>2)*16 ; cols c0 = (w&3)*16
// f32 WMMA fragment layouts per ISA 7.12.2:
//   A 16x4 : lane {m=lane&15, half=lane>>4}: v0=A[m][2h], v1=A[m][2h+1]
//   B 4x16 : lane {n=lane&15, half}:         v0=B[2h][n], v1=B[2h+1][n]
//   C 16x16: reg v, lane: row = v + 8*half, col = lane&15
// -------------------------------------------------------------------------
__global__ __launch_bounds__(256) void gat_gemm1(const float* __restrict__ x,
                                                 const float* __restrict__ W,
                                                 float* __restrict__ Wh) {
    const int h    = blockIdx.y;
    const int wave = threadIdx.x >> 5;
    const int lane = threadIdx.x & 31;
    const int half = lane >> 4;
    const int nl   = lane & 15;

    const int m0 = blockIdx.x * 32 + (wave >> 2) * 16;
    const int c0 = (wave & 3) * 16;

    const float* Whead = W + (size_t)h * GIN * GO;
    const int arow = m0 + nl;

    v8f acc = {};
#pragma unroll 4
    for (int k = 0; k < GIN; k += 4) {
        const int kr = k + 2 * half;
        v2f a = *(const v2f*)(x + (size_t)arow * GIN + kr);   // 8B aligned
        v2f b;
        b.x = Whead[(size_t)kr * GO + c0 + nl];
        b.y = Whead[(size_t)(kr + 1) * GO + c0 + nl];
        acc = __builtin_amdgcn_wmma_f32_16x16x4_f32(
            false, a, false, b, (short)0, acc, false, false);
    }

    float* o = Wh + (size_t)h * GN * GO;
#pragma unroll
    for (int v = 0; v < 8; ++v) {
        const int r = m0 + v + 8 * half;
        o[(size_t)r * GO + c0 + nl] = acc[v];
    }
}

// -------------------------------------------------------------------------
// Kernel 2: f1[h,n] = Wh[h,n,:]·a1[h,:], f2 likewise. One wave32 per row.
// -------------------------------------------------------------------------
__global__ __launch_bounds__(256) void gat_f12(const float* __restrict__ Wh,
                                               const float* __restrict__ a1,
                                               const float* __restrict__ a2,
                                               float* __restrict__ f1,
                                               float* __restrict__ f2) {
    const int wave = threadIdx.x >> 5;
    const int lane = threadIdx.x & 31;
    const int r = blockIdx.x * 8 + wave;          // flat h*N + n, < H*N
    const int h = r / GN;

    const float* row = Wh + (size_t)r * GO;
    const float w0 = row[lane];
    const float w1 = row[32 + lane];
    float s1 = w0 * a1[h * GO + lane] + w1 * a1[h * GO + 32 + lane];
    float s2 = w0 * a2[h * GO + lane] + w1 * a2[h * GO + 32 + lane];
#pragma unroll
    for (int off = 16; off > 0; off >>= 1) {
        s1 += __shfl_xor(s1, off, 32);
        s2 += __shfl_xor(s2, off, 32);
    }
    if (lane == 0) { f1[r] = s1; f2[r] = s2; }
}

// -------------------------------------------------------------------------
// Kernel 3: fused masked softmax + attention@Wh + ELU.
// Block = 128 threads (4 waves) handles 16 rows of one head.
// Online softmax (flash style): running max m / sum l live in registers of
// threads 0..15 (wave 0); accumulator acc[16x64] lives in WMMA C fragments,
// one 16x16 column tile per wave. Per 16-wide j chunk:
//   e-tile -> LDS, stats+P in LDS, rescale acc, 4x wmma_f32_16x16x4.
// -------------------------------------------------------------------------
__global__ __launch_bounds__(128) void gat_attn(const float* __restrict__ f1,
                                                const float* __restrict__ f2,
                                                const int*   __restrict__ adj,
                                                const float* __restrict__ Wh,
                                                float* __restrict__ out) {
    __shared__ float P[16][17];    // score tile, then probability tile
    __shared__ float fac[16];      // per-row rescale factor / final 1/l
    __shared__ float f1s[16];

    const int h    = blockIdx.y;
    const int i0   = blockIdx.x * 16;
    const int tid  = threadIdx.x;
    const int wave = tid >> 5;
    const int lane = tid & 31;
    const int half = lane >> 4;
    const int nl   = lane & 15;
    const int c0   = wave * 16;    // this wave's 16 output columns

    const float* f2h  = f2 + (size_t)h * GN;
    const float* Whh  = Wh + (size_t)h * GN * GO;

    if (tid < 16) f1s[tid] = f1[(size_t)h * GN + i0 + tid];

    v8f acc = {};
    float mrow = -INFINITY;   // valid in threads 0..15 only
    float lrow = 0.0f;

    __syncthreads();

    for (int j0 = 0; j0 < GN; j0 += 16) {
        // ---- step 1: masked leaky-relu score tile (all 128 threads) ----
#pragma unroll
        for (int t = tid; t < 256; t += 128) {
            const int r = t >> 4, c = t & 15;
            const int i = i0 + r, j = j0 + c;
            float s = f1s[r] + f2h[j];
            s = (s > 0.0f) ? s : GALPHA * s;
            if (adj[(size_t)i * GN + j] <= 0) s = GNEGINF;
            P[r][c] = s;
        }
        __syncthreads();

        // ---- step 2: online-softmax stats, row-owned by threads 0..15 ----
        if (tid < 16) {
            float cmax = P[tid][0];
#pragma unroll
            for (int c = 1; c < 16; ++c) cmax = fmaxf(cmax, P[tid][c]);
            const float nm = fmaxf(mrow, cmax);
            const float f  = expf(mrow - nm);   // exp(-inf)=0 on first chunk
            float psum = 0.0f;
#pragma unroll
            for (int c = 0; c < 16; ++c) {
                const float p = expf(P[tid][c] - nm);
                P[tid][c] = p;
                psum += p;
            }
            lrow = lrow * f + psum;
            mrow = nm;
            fac[tid] = f;
        }
        __syncthreads();

        // ---- step 3: rescale accumulator, then P(16x16) @ Wh(16x64) ----
#pragma unroll
        for (int v = 0; v < 8; ++v) acc[v] *= fac[v + 8 * half];

#pragma unroll
        for (int kk = 0; kk < 16; kk += 4) {
            v2f a;
            a.x = P[nl][kk + 2 * half];
            a.y = P[nl][kk + 2 * half + 1];
            v2f b;
            const int jr = j0 + kk + 2 * half;
            b.x = Whh[(size_t)jr * GO + c0 + nl];
            b.y = Whh[(size_t)(jr + 1) * GO + c0 + nl];
            acc = __builtin_amdgcn_wmma_f32_16x16x4_f32(
                false, a, false, b, (short)0, acc, false, false);
        }
        __syncthreads();   // protect P before next chunk overwrites it
    }

    // ---- epilogue: normalize, ELU, store concat-heads layout ----
    if (tid < 16) fac[tid] = 1.0f / lrow;
    __syncthreads();

#pragma unroll
    for (int v = 0; v < 8; ++v) {
        const int r = v + 8 * half;
        float val = acc[v] * fac[r];
        val = (val > 0.0f) ? val : expm1f(val);
        out[(size_t)(i0 + r) * (GH * GO) + h * GO + c0 + nl] = val;
    }
}

// -------------------------------------------------------------------------
extern "C" void kernel_launch(void* const* d_in, const int* in_sizes, int n_in,
                              void* d_out, int out_size, void* d_ws, size_t ws_size,
                              hipStream_t stream) {
    const float* x   = (const float*)d_in[0];
    const int*   adj = (const int*)  d_in[1];
    const float* W   = (const float*)d_in[2];
    const float* a1  = (const float*)d_in[3];
    const float* a2  = (const float*)d_in[4];
    float* out = (float*)d_out;

    // workspace: Wh [H,N,64] f32 (6.3MB), f1 [H,N], f2 [H,N]
    float* Wh = (float*)d_ws;
    float* f1 = Wh + (size_t)GH * GN * GO;
    float* f2 = f1 + (size_t)GH * GN;

    gat_gemm1<<<dim3(GN / 32, GH), 256, 0, stream>>>(x, W, Wh);
    gat_f12  <<<(GH * GN) / 8,     256, 0, stream>>>(Wh, a1, a2, f1, f2);
    gat_attn <<<dim3(GN / 16, GH), 128, 0, stream>>>(f1, f2, adj, Wh, out);
}